// EnhancedConsistencyLoss_28948079575214
// MI455X (gfx1250) — compile-verified
//
#include <hip/hip_runtime.h>

typedef __attribute__((ext_vector_type(2))) float v2f;
typedef __attribute__((ext_vector_type(8))) float v8f;

#define NUM_E   8
#define DIM     128
#define PAIRS   28
#define BETA_C  0.1f

__device__ __forceinline__ float wave_reduce_sum(float v) {
#pragma unroll
    for (int m = 16; m >= 1; m >>= 1) v += __shfl_xor(v, m, 32);
    return v;
}
__device__ __forceinline__ float wave_reduce_max(float v) {
#pragma unroll
    for (int m = 16; m >= 1; m >>= 1) v = fmaxf(v, __shfl_xor(v, m, 32));
    return v;
}

// One workgroup (8 wave32 waves) per ranked node.
// Wave e: log-softmax of ranked row e; then wave 0 computes the 8x8 Gram
// G = U * LP^T with fp32 WMMA (v_wmma_f32_16x16x4_f32), accumulating K=128.
__global__ __launch_bounds__(256) void ecl_node_kernel(
    const float* __restrict__ expert_outputs,
    const long long* __restrict__ rankings,
    const long long* __restrict__ node_indices,
    float* __restrict__ partial, int S)
{
    __shared__ float lds_u [16][DIM];  // rows 8..15 zero-padded for 16x16 WMMA tile
    __shared__ float lds_lp[16][DIM];
    __shared__ float sA[NUM_E];

    const int s    = blockIdx.x;
    const int w    = threadIdx.x >> 5;   // wave id == ranked-expert slot (0..7)
    const int lane = threadIdx.x & 31;
    if (s >= S) return;

    const long long node = node_indices[s];
    const long long re   = rankings[(long long)s * NUM_E + w];
    const float* row = expert_outputs + (node * NUM_E + re) * DIM;

    // 4 contiguous floats per lane -> coalesced global_load_b128 per wave
    const float4 x = *(const float4*)(row + lane * 4);

    float m = fmaxf(fmaxf(x.x, x.y), fmaxf(x.z, x.w));
    m = wave_reduce_max(m);
    const float e0 = __expf(x.x - m), e1 = __expf(x.y - m),
                e2 = __expf(x.z - m), e3 = __expf(x.w - m);
    const float ssum = wave_reduce_sum(e0 + e1 + e2 + e3);
    const float lse  = m + __logf(ssum);

    float aacc = 0.f;
    const int base = lane * 4;
    const float xv[4] = {x.x, x.y, x.z, x.w};
#pragma unroll
    for (int i = 0; i < 4; ++i) {
        const float lp = xv[i] - lse;       // log_softmax
        const float p  = __expf(lp);        // prob
        const float u  = __expf(p);         // exp(prob)  (faithful to reference)
        aacc += u * p;                      // diag term A_e
        lds_lp[w][base + i]     = lp;
        lds_u [w][base + i]     = u;
        lds_lp[w + 8][base + i] = 0.f;      // zero-pad rows 8..15
        lds_u [w + 8][base + i] = 0.f;
    }
    aacc = wave_reduce_sum(aacc);
    if (lane == 0) sA[w] = aacc;

    __syncthreads();

    if (w == 0) {  // wave-uniform branch: EXEC all ones inside (WMMA requirement)
        // fp32 WMMA operand layout, A(16x4) and B(4x16), K-chunk base kb:
        //   lane L: row r = L%16, columns (kb + 2*(L/16)) and +1, for BOTH A and B.
        const int r   = lane & 15;
        const int grp = lane >> 4;
        v8f c = {0.f, 0.f, 0.f, 0.f, 0.f, 0.f, 0.f, 0.f};
#pragma unroll
        for (int kb = 0; kb < DIM; kb += 4) {
            const int col = kb + 2 * grp;
            v2f a, b;
            a[0] = lds_u [r][col]; a[1] = lds_u [r][col + 1];
            b[0] = lds_lp[r][col]; b[1] = lds_lp[r][col + 1];
            c = __builtin_amdgcn_wmma_f32_16x16x4_f32(
                    false, a, false, b, (short)0, c, false, false);
        }
        // Rows/cols 8..15 of G are exactly zero, so sum(all C elems) = sum_{j,k<8} G.
        float tot = c[0] + c[1] + c[2] + c[3] + c[4] + c[5] + c[6] + c[7];
        // trace: G[j][j] sits at lane j (N=j), VGPR j (M=j)
        float diag = 0.f;
#pragma unroll
        for (int i = 0; i < 8; ++i) diag = (lane == i) ? c[i] : diag;
        tot  = wave_reduce_sum(tot);
        diag = wave_reduce_sum(diag);
        if (lane == 0) {
            float sumA = 0.f;
#pragma unroll
            for (int i = 0; i < NUM_E; ++i) sumA += sA[i];
            // sum over 28 pairs of 0.5*(t1+t2) = 0.5*(7*sum A_e - offdiag(G))
            partial[s] = 0.5f * (7.0f * sumA - (tot - diag));
        }
    }
}

// Deterministic fixed-order final reduction (no float atomics).
__global__ __launch_bounds__(256) void ecl_reduce_kernel(
    const float* __restrict__ partial, float* __restrict__ out, int S)
{
    __shared__ float buf[256];
    float sum = 0.f;
    for (int i = threadIdx.x; i < S; i += 256) sum += partial[i];
    buf[threadIdx.x] = sum;
    __syncthreads();
    for (int half = 128; half > 0; half >>= 1) {
        if ((int)threadIdx.x < half) buf[threadIdx.x] += buf[threadIdx.x + half];
        __syncthreads();
    }
    if (threadIdx.x == 0)
        out[0] = BETA_C * buf[0] / ((float)S * (float)PAIRS);
}

extern "C" void kernel_launch(void* const* d_in, const int* in_sizes, int n_in,
                              void* d_out, int out_size, void* d_ws, size_t ws_size,
                              hipStream_t stream) {
    const float*     expert_outputs = (const float*)d_in[0];
    const long long* rankings       = (const long long*)d_in[1];  // int64
    const long long* node_indices   = (const long long*)d_in[2];  // int64
    const int S = in_sizes[2];                                    // NUM_RANKED

    float* partial = (float*)d_ws;  // S floats of scratch

    ecl_node_kernel<<<S, 256, 0, stream>>>(expert_outputs, rankings, node_indices,
                                           partial, S);
    ecl_reduce_kernel<<<1, 256, 0, stream>>>(partial, (float*)d_out, S);
}